// ComplexMixture_34583076667832
// MI455X (gfx1250) — compile-verified
//
#include <hip/hip_runtime.h>
#include <cstdint>

typedef float v2f __attribute__((ext_vector_type(2)));
typedef float v8f __attribute__((ext_vector_type(8)));

#define BATCH 32
#define SEQ 8192
#define DIM 64
#define CHUNKS 8
#define CS (SEQ / CHUNKS)   // 1024 seq rows per block
#define SR 64               // seq rows staged in LDS per buffer
#define LDS_STRIDE 80       // 64 + 16 pad: lanes 0-15 vs 16-31 hit disjoint banks

#if __has_builtin(__builtin_amdgcn_global_load_async_to_lds_b128) && \
    __has_builtin(__builtin_amdgcn_s_wait_asynccnt)
#define HAS_ASYNC_LDS 1
#else
#define HAS_ASYNC_LDS 0
#endif

#if HAS_ASYNC_LDS
// Builtin expects: (int4 AS1* gsrc, int4 AS3* ldst, i32 offset, i32 cpol)
// (clang prints AS1 as `__device__` in its diagnostics).
typedef int v4i_vs __attribute__((vector_size(16)));
typedef __attribute__((address_space(1))) v4i_vs* gas1_ptr;
typedef __attribute__((address_space(3))) v4i_vs* las3_ptr;
__device__ __forceinline__ void async_copy16(const float* g, float* l) {
    // 16B global -> LDS via the CDNA5 async path (ASYNCcnt-tracked DMA).
    // Generic LDS pointers carry the LDS byte offset in their low 32 bits.
    __builtin_amdgcn_global_load_async_to_lds_b128(
        (gas1_ptr)(uintptr_t)g, (las3_ptr)(uint32_t)(uintptr_t)l, 0, 0);
}
#endif

// Fragment of a [rows x 64] fp32 tile held in LDS, in the WMMA 16x16x4 f32
// A/B operand layout: lanes 0-15 hold K=0 (v.x) / K=1 (v.y), lanes 16-31 hold
// K=2 / K=3, column (M or N) = lane & 15 within the 16-wide feature tile.
__device__ __forceinline__ v2f load_frag(const float* lds, int r0, int tile, int lane) {
    const int col = tile * 16 + (lane & 15);
    const int row = r0 + ((lane >> 4) << 1);
    v2f f;
    f.x = lds[row * LDS_STRIDE + col];
    f.y = lds[(row + 1) * LDS_STRIDE + col];
    return f;
}

// One SR-row stage worth of WMMA work for this wave's 2x2 tile block.
__device__ __forceinline__ void compute_stage(const float* lr, const float* li,
                                              int lane, int i0, int j0,
                                              v8f accR[2][2], v8f accX[2][2]) {
    #pragma unroll 2
    for (int r0 = 0; r0 < SR; r0 += 4) {
        v2f ar[2], ai[2], br[2], bi[2];
        ar[0] = load_frag(lr, r0, i0,     lane);
        ar[1] = load_frag(lr, r0, i0 + 1, lane);
        ai[0] = load_frag(li, r0, i0,     lane);
        ai[1] = load_frag(li, r0, i0 + 1, lane);
        br[0] = load_frag(lr, r0, j0,     lane);
        br[1] = load_frag(lr, r0, j0 + 1, lane);
        bi[0] = load_frag(li, r0, j0,     lane);
        bi[1] = load_frag(li, r0, j0 + 1, lane);
        #pragma unroll
        for (int a = 0; a < 2; ++a) {
            #pragma unroll
            for (int c = 0; c < 2; ++c) {
                // out_real partial: rr then ii chained into one C accumulator
                accR[a][c] = __builtin_amdgcn_wmma_f32_16x16x4_f32(
                    false, ar[a], false, br[c], (short)0, accR[a][c], false, false);
                accR[a][c] = __builtin_amdgcn_wmma_f32_16x16x4_f32(
                    false, ai[a], false, bi[c], (short)0, accR[a][c], false, false);
                // ri partial
                accX[a][c] = __builtin_amdgcn_wmma_f32_16x16x4_f32(
                    false, ar[a], false, bi[c], (short)0, accX[a][c], false, false);
            }
        }
    }
}

__global__ __launch_bounds__(128) void gram_partial_kernel(
    const float* __restrict__ real, const float* __restrict__ imag,
    float* __restrict__ ws) {
#if HAS_ASYNC_LDS
    __shared__ float lr[2][SR * LDS_STRIDE];
    __shared__ float li[2][SR * LDS_STRIDE];
#else
    __shared__ float lr[1][SR * LDS_STRIDE];
    __shared__ float li[1][SR * LDS_STRIDE];
#endif

    const int b     = blockIdx.y;
    const int chunk = blockIdx.x;
    const int tid   = threadIdx.x;
    const int lane  = tid & 31;
    const int wave  = tid >> 5;

    // 4 waves, each owns a 2x2 block of 16x16 tiles of the 64x64 output
    const int i0 = (wave >> 1) * 2;
    const int j0 = (wave & 1) * 2;

    const float* Rg = real + ((size_t)b * SEQ + (size_t)chunk * CS) * DIM;
    const float* Ig = imag + ((size_t)b * SEQ + (size_t)chunk * CS) * DIM;

    const v8f vzero = {0.f, 0.f, 0.f, 0.f, 0.f, 0.f, 0.f, 0.f};
    v8f accR[2][2]; // rr + ii accumulator (chained)
    v8f accX[2][2]; // ri accumulator
    for (int a = 0; a < 2; ++a)
        for (int c = 0; c < 2; ++c) { accR[a][c] = vzero; accX[a][c] = vzero; }

    constexpr int NSTAGE = CS / SR;

#if HAS_ASYNC_LDS
    // Prologue: async-prefetch stage 0 into buffer 0.
    {
        #pragma unroll
        for (int q = 0; q < 8; ++q) {
            const int flat = tid * 8 + q;       // 0..1023 float4 slots
            const int row  = flat >> 4;
            const int c4   = (flat & 15) << 2;
            async_copy16(Rg + flat * 4, &lr[0][row * LDS_STRIDE + c4]);
            async_copy16(Ig + flat * 4, &li[0][row * LDS_STRIDE + c4]);
        }
    }
    for (int stage = 0; stage < NSTAGE; ++stage) {
        const int cur = stage & 1;
        // Publish buffer `cur`: my async copies done, then all waves' copies done.
        __builtin_amdgcn_s_wait_asynccnt(0);
        __syncthreads();
        // Prefetch next stage into the other buffer (safe: the barrier above
        // proves every wave finished computing on it last iteration).
        if (stage + 1 < NSTAGE) {
            const float* Rn = Rg + (size_t)(stage + 1) * SR * DIM;
            const float* In = Ig + (size_t)(stage + 1) * SR * DIM;
            #pragma unroll
            for (int q = 0; q < 8; ++q) {
                const int flat = tid * 8 + q;
                const int row  = flat >> 4;
                const int c4   = (flat & 15) << 2;
                async_copy16(Rn + flat * 4, &lr[1 - cur][row * LDS_STRIDE + c4]);
                async_copy16(In + flat * 4, &li[1 - cur][row * LDS_STRIDE + c4]);
            }
        }
        compute_stage(lr[cur], li[cur], lane, i0, j0, accR, accX);
    }
#else
    for (int stage = 0; stage < NSTAGE; ++stage) {
        const float4* Rg4 = (const float4*)(Rg + (size_t)stage * SR * DIM);
        const float4* Ig4 = (const float4*)(Ig + (size_t)stage * SR * DIM);
        #pragma unroll
        for (int q = 0; q < 8; ++q) {
            const int flat = tid * 8 + q;
            const int row  = flat >> 4;
            const int c4   = (flat & 15) << 2;
            *(float4*)&lr[0][row * LDS_STRIDE + c4] = Rg4[flat];
            *(float4*)&li[0][row * LDS_STRIDE + c4] = Ig4[flat];
        }
        __syncthreads();
        compute_stage(lr[0], li[0], lane, i0, j0, accR, accX);
        __syncthreads();
    }
#endif

    // Write per-chunk partials: ws[((chunk*B + b)*2 + {0:real,1:ri})*4096 + r*64 + c]
    float* wsR = ws + ((size_t)(chunk * BATCH + b) * 2 + 0) * (DIM * DIM);
    float* wsX = ws + ((size_t)(chunk * BATCH + b) * 2 + 1) * (DIM * DIM);
    #pragma unroll
    for (int a = 0; a < 2; ++a) {
        #pragma unroll
        for (int c = 0; c < 2; ++c) {
            #pragma unroll
            for (int v = 0; v < 8; ++v) {
                // C/D layout: VGPR v, lanes 0-15 -> M=v, lanes 16-31 -> M=v+8
                const int row = (i0 + a) * 16 + v + ((lane >> 4) << 3);
                const int col = (j0 + c) * 16 + (lane & 15);
                wsR[row * DIM + col] = accR[a][c][v];
                wsX[row * DIM + col] = accX[a][c][v];
            }
        }
    }
}

__global__ __launch_bounds__(256) void finalize_kernel(
    const float* __restrict__ ws, float* __restrict__ out) {
    const int idx = blockIdx.x * blockDim.x + threadIdx.x; // over B*64*64
    if (idx >= BATCH * DIM * DIM) return;
    const int b  = idx >> 12;
    const int rc = idx & 4095;
    const int r  = rc >> 6;
    const int c  = rc & 63;
    float sR = 0.f, s1 = 0.f, s2 = 0.f;
    for (int ch = 0; ch < CHUNKS; ++ch) {
        const float* base = ws + ((size_t)(ch * BATCH + b) * 2) * (DIM * DIM);
        sR += base[rc];                       // rr+ii partial
        s1 += base[DIM * DIM + r * DIM + c];  // ri[r,c]
        s2 += base[DIM * DIM + c * DIM + r];  // ri[c,r]
    }
    const float inv_s = 1.0f / (float)SEQ;
    out[(size_t)b * (DIM * DIM) + rc] = sR * inv_s;                   // out_real
    out[(size_t)(BATCH + b) * (DIM * DIM) + rc] = (s1 - s2) * inv_s;  // out_imag
}

extern "C" void kernel_launch(void* const* d_in, const int* in_sizes, int n_in,
                              void* d_out, int out_size, void* d_ws, size_t ws_size,
                              hipStream_t stream) {
    (void)in_sizes; (void)n_in; (void)out_size; (void)ws_size;
    const float* real = (const float*)d_in[0];
    const float* imag = (const float*)d_in[1];
    float* out = (float*)d_out;
    float* ws  = (float*)d_ws;   // needs CHUNKS*BATCH*2*4096*4 = 8 MB

    dim3 grid(CHUNKS, BATCH);
    gram_partial_kernel<<<grid, 128, 0, stream>>>(real, imag, ws);

    const int n = BATCH * DIM * DIM;
    finalize_kernel<<<(n + 255) / 256, 256, 0, stream>>>(ws, out);
}